// PillarFeatureNet_88313117540619
// MI455X (gfx1250) — compile-verified
//
#include <hip/hip_runtime.h>

typedef float v2f __attribute__((ext_vector_type(2)));
typedef float v8f __attribute__((ext_vector_type(8)));

#define PTS 32
#define CH 4
#define UNITS 64
#define LDS_STRIDE 13   // 12 K-values padded to 13 to avoid LDS bank conflicts
#define WAVES_PER_BLOCK 8

__global__ __launch_bounds__(256) void pfn_wmma_kernel(
    const float* __restrict__ features,   // [N,32,4]
    const int*   __restrict__ num_voxels, // [N]
    const int*   __restrict__ coors,      // [N,4]
    const float* __restrict__ Wm,         // [64,9]
    const float* __restrict__ gammav,     // [64]
    const float* __restrict__ betav,      // [64]
    const float* __restrict__ rmean,      // [64]
    const float* __restrict__ rvar,       // [64]
    float* __restrict__ out,              // [N,64]
    int n_pillars)
{
    __shared__ float sbuf[WAVES_PER_BLOCK][PTS * LDS_STRIDE];

    const int lane  = threadIdx.x & 31;
    const int wave  = threadIdx.x >> 5;
    const int pillar = blockIdx.x * WAVES_PER_BLOCK + wave;

    const int col   = lane & 15;   // N column within a 16-wide tile (also M row for A)
    const int khalf = lane >> 4;   // which K pair this lane holds (A/B f32 layout)

    // ---- Preload B fragments (W, K x N layout: B[k][u] = W[u][k]) + folded BN affine.
    // B 4x16 f32 fragment: lane holds column u = nt*16+col, K pair = khalf*2 + {0,1}.
    v2f  Bf[4][3];
    float aff_a[4], aff_b[4];
#pragma unroll
    for (int nt = 0; nt < 4; ++nt) {
        const int u = nt * 16 + col;
#pragma unroll
        for (int kt = 0; kt < 3; ++kt) {
            const int k0 = kt * 4 + khalf * 2;
            v2f b;
            b[0] = (k0     < 9) ? Wm[u * 9 + k0]     : 0.0f;  // K=9..11 zero pad
            b[1] = (k0 + 1 < 9) ? Wm[u * 9 + k0 + 1] : 0.0f;
            Bf[nt][kt] = b;
        }
        const float s = gammav[u] * rsqrtf(rvar[u] + 1e-3f);
        aff_a[nt] = s;
        aff_b[nt] = betav[u] - rmean[u] * s;
    }

    const bool active = (pillar < n_pillars);   // wave-uniform: EXEC stays full inside

    if (active) {
        // ---- Stage 1: lane p builds the 9-dim (padded to 12) feature of point p.
        const float4 f = *(const float4*)(features + (size_t)pillar * PTS * CH + lane * CH);

        // Pillar mean over ALL 32 points (reference sums full P, divides by nv).
        float sx = f.x, sy = f.y, sz = f.z;
#pragma unroll
        for (int m = 16; m >= 1; m >>= 1) {
            sx += __shfl_xor(sx, m, 32);
            sy += __shfl_xor(sy, m, 32);
            sz += __shfl_xor(sz, m, 32);
        }
        const int   nv  = num_voxels[pillar];
        const float inv = 1.0f / (float)nv;
        const float mx = sx * inv, my = sy * inv, mz = sz * inv;

        const float cx = (float)coors[pillar * 4 + 3] * 0.2f + 0.1f;           // VX/2 + 0.0
        const float cy = (float)coors[pillar * 4 + 2] * 0.2f + (0.1f - 40.0f); // VY/2 - 40.0

        const float msk = (lane < nv) ? 1.0f : 0.0f;

        float* row = &sbuf[wave][lane * LDS_STRIDE];
        row[0]  = f.x * msk;
        row[1]  = f.y * msk;
        row[2]  = f.z * msk;
        row[3]  = f.w * msk;
        row[4]  = (f.x - mx) * msk;
        row[5]  = (f.y - my) * msk;
        row[6]  = (f.z - mz) * msk;
        row[7]  = (f.x - cx) * msk;
        row[8]  = (f.y - cy) * msk;
        row[9]  = 0.0f;
        row[10] = 0.0f;
        row[11] = 0.0f;
    }
    __syncthreads();   // make LDS rows visible across lanes (and order the ds ops)

    if (active) {
        // ---- Stage 2: gather A fragments per ISA 16x4 f32 A layout.
        // Lane holds row M = col of tile m (point p = m*16+col), K pair khalf*2+{0,1}.
        v2f Af[2][3];
#pragma unroll
        for (int m = 0; m < 2; ++m) {
            const float* arow = &sbuf[wave][(m * 16 + col) * LDS_STRIDE];
#pragma unroll
            for (int kt = 0; kt < 3; ++kt) {
                const int k0 = kt * 4 + khalf * 2;
                v2f a;
                a[0] = arow[k0];
                a[1] = arow[k0 + 1];
                Af[m][kt] = a;
            }
        }

        v8f acc[2][4];
#pragma unroll
        for (int m = 0; m < 2; ++m)
#pragma unroll
            for (int nt = 0; nt < 4; ++nt) {
                v8f z = {0.0f, 0.0f, 0.0f, 0.0f, 0.0f, 0.0f, 0.0f, 0.0f};
                acc[m][nt] = z;
            }

        // ---- Stage 3: 24x V_WMMA_F32_16X16X4_F32, D = A*B + C over 3 K-steps.
#pragma unroll
        for (int kt = 0; kt < 3; ++kt)
#pragma unroll
            for (int m = 0; m < 2; ++m)
#pragma unroll
                for (int nt = 0; nt < 4; ++nt)
                    acc[m][nt] = __builtin_amdgcn_wmma_f32_16x16x4_f32(
                        /*neg_a=*/false, Af[m][kt],
                        /*neg_b=*/false, Bf[nt][kt],
                        /*c_mod=*/(short)0, acc[m][nt],
                        /*reuse_a=*/false, /*reuse_b=*/false);

        // ---- Stage 4: fused BN-affine + ReLU + max over 32 points.
        // D layout: lane holds N=col, rows M = khalf*8 + r (r=0..7) of tile m.
#pragma unroll
        for (int nt = 0; nt < 4; ++nt) {
            float vmax = 0.0f;   // ReLU floor: max_p relu(y) == max(0, max_p y)
#pragma unroll
            for (int m = 0; m < 2; ++m) {
                v8f d = acc[m][nt];
#pragma unroll
                for (int r = 0; r < 8; ++r)
                    vmax = fmaxf(vmax, fmaf(d[r], aff_a[nt], aff_b[nt]));
            }
            vmax = fmaxf(vmax, __shfl_xor(vmax, 16, 32));  // combine the two row halves
            if (lane < 16)
                out[(size_t)pillar * UNITS + nt * 16 + lane] = vmax;
        }
    }
}

extern "C" void kernel_launch(void* const* d_in, const int* in_sizes, int n_in,
                              void* d_out, int out_size, void* d_ws, size_t ws_size,
                              hipStream_t stream) {
    (void)n_in; (void)out_size; (void)d_ws; (void)ws_size;
    const float* features   = (const float*)d_in[0];
    const int*   num_voxels = (const int*)  d_in[1];
    const int*   coors      = (const int*)  d_in[2];
    const float* Wm         = (const float*)d_in[3];
    const float* gammav     = (const float*)d_in[4];
    const float* betav      = (const float*)d_in[5];
    const float* rmean      = (const float*)d_in[6];
    const float* rvar       = (const float*)d_in[7];
    float* out = (float*)d_out;

    const int n_pillars = in_sizes[1];                 // num_voxels is [N]
    const int blocks = (n_pillars + WAVES_PER_BLOCK - 1) / WAVES_PER_BLOCK;
    pfn_wmma_kernel<<<blocks, 256, 0, stream>>>(
        features, num_voxels, coors, Wm, gammav, betav, rmean, rvar, out, n_pillars);
}